// E_EdgeConv_67267777790372
// MI455X (gfx1250) — compile-verified
//
#include <hip/hip_runtime.h>
#include <hip/hip_bf16.h>

typedef __attribute__((ext_vector_type(2))) float v2f;
typedef __attribute__((ext_vector_type(8))) float v8f;

#ifndef NB_MEAN
#define NB_MEAN 1024
#endif
// workspace layout (floats): [0..2] = global sum of h[edge_dst]; [16..16+3*NB_MEAN) = block partials
#define WS_SUMS      0
#define WS_PARTIALS  16

// ---------------------------------------------------------------------------
// Kernel 1: zero the output accumulator (harness poisons d_out before timing)
// ---------------------------------------------------------------------------
__global__ void ec_zero_kernel(float* __restrict__ out, int n) {
    int i = blockIdx.x * blockDim.x + threadIdx.x;
    if (i < n) out[i] = 0.0f;
}

// ---------------------------------------------------------------------------
// Kernel 2: per-block partial sums of h[edge_dst] (deterministic tree reduce)
// ---------------------------------------------------------------------------
__global__ __launch_bounds__(256)
void ec_mean_partial_kernel(const float* __restrict__ h,
                            const int* __restrict__ edst,
                            float* __restrict__ partials, int E) {
    __shared__ float red0[256], red1[256], red2[256];
    float sx = 0.f, sy = 0.f, sz = 0.f;
    int stride = gridDim.x * blockDim.x;
    for (int e = blockIdx.x * blockDim.x + threadIdx.x; e < E; e += stride) {
        int d = edst[e];
        const float* hp = h + 3 * (long)d;
        sx += hp[0]; sy += hp[1]; sz += hp[2];
    }
    red0[threadIdx.x] = sx; red1[threadIdx.x] = sy; red2[threadIdx.x] = sz;
    __syncthreads();
    for (int off = 128; off > 0; off >>= 1) {
        if ((int)threadIdx.x < off) {
            red0[threadIdx.x] += red0[threadIdx.x + off];
            red1[threadIdx.x] += red1[threadIdx.x + off];
            red2[threadIdx.x] += red2[threadIdx.x + off];
        }
        __syncthreads();
    }
    if (threadIdx.x == 0) {
        partials[blockIdx.x * 3 + 0] = red0[0];
        partials[blockIdx.x * 3 + 1] = red1[0];
        partials[blockIdx.x * 3 + 2] = red2[0];
    }
}

// ---------------------------------------------------------------------------
// Kernel 3: reduce block partials -> global sums (single block, deterministic)
// ---------------------------------------------------------------------------
__global__ __launch_bounds__(256)
void ec_mean_final_kernel(const float* __restrict__ partials,
                          float* __restrict__ sums, int nb) {
    __shared__ float red0[256], red1[256], red2[256];
    float sx = 0.f, sy = 0.f, sz = 0.f;
    for (int i = threadIdx.x; i < nb; i += 256) {
        sx += partials[i * 3 + 0];
        sy += partials[i * 3 + 1];
        sz += partials[i * 3 + 2];
    }
    red0[threadIdx.x] = sx; red1[threadIdx.x] = sy; red2[threadIdx.x] = sz;
    __syncthreads();
    for (int off = 128; off > 0; off >>= 1) {
        if ((int)threadIdx.x < off) {
            red0[threadIdx.x] += red0[threadIdx.x + off];
            red1[threadIdx.x] += red1[threadIdx.x + off];
            red2[threadIdx.x] += red2[threadIdx.x + off];
        }
        __syncthreads();
    }
    if (threadIdx.x == 0) { sums[0] = red0[0]; sums[1] = red1[0]; sums[2] = red2[0]; }
}

// ---------------------------------------------------------------------------
// Kernel 4: fused gather -> features -> 7x7 linear via V_WMMA_F32_16X16X4_F32
//           -> f32 atomic scatter-add by edge_dst.
// Each wave32 handles 32 edges = two 16-edge WMMA tiles.
//   A (16x4, f32) = W rows (M = output feature j, wave-invariant)
//   B (4x16, f32) = feat^T (K = input feature, N = edge), built via shuffles
//   C (16x16 f32) = bias broadcast; K=7 padded to 8 -> two chained k=4 WMMAs
// D layout: lane L<16 holds e[edge_L][j] in acc[j], j=0..6 -> 7 atomics/lane.
// ---------------------------------------------------------------------------
__device__ __forceinline__ float ec_wload(const float* __restrict__ W, int j, int k) {
    return (j < 7 && k < 7) ? W[j * 7 + k] : 0.0f;
}

__global__ __launch_bounds__(256)
void ec_edge_wmma_kernel(const float* __restrict__ h,
                         const int* __restrict__ esrc,
                         const int* __restrict__ edst,
                         const float* __restrict__ W,
                         const float* __restrict__ b,
                         const float* __restrict__ sums,
                         float* __restrict__ out, int E) {
    const int lane   = threadIdx.x & 31;
    const int waveId = (blockIdx.x * blockDim.x + threadIdx.x) >> 5;
    const long base  = (long)waveId * 32;

    const float invE  = 1.0f / (float)E;
    const float rsqE  = rsqrtf((float)E);
    const float mx = sums[0] * invE, my = sums[1] * invE, mz = sums[2] * invE;

    // --- per-lane edge features (clamped for tail; EXEC stays full) ---
    const long e   = base + lane;
    const bool act = (e < E);
    const long ec  = act ? e : 0;
    const int  s   = esrc[ec];
    const int  d   = edst[ec];
    const float* hs = h + 3 * (long)s;
    const float* hd = h + 3 * (long)d;
    float sx = hs[0], sy = hs[1], sz = hs[2];
    float dx = hd[0], dy = hd[1], dz = hd[2];

    float f0 = dx - sx, f1 = dy - sy, f2 = dz - sz;          // delta = dst - src
    float f6 = sqrtf(f0 * f0 + f1 * f1 + f2 * f2);           // dist
    float f3 = fabsf(sx - mx) * rsqE;                        // std_xyt
    float f4 = fabsf(sy - my) * rsqE;
    float f5 = fabsf(sz - mz) * rsqE;
    float f7 = 0.0f;                                         // K padding
    if (!act) { f0 = f1 = f2 = f3 = f4 = f5 = f6 = 0.0f; }

    // --- A = W, laid out per 32-bit 16x4 A-matrix spec (wave-invariant) ---
    const int m  = lane & 15;        // M = output feature row
    const int kh = lane >> 4;        // lane half selects K pair
    v2f A0, A1;
    A0[0] = ec_wload(W, m, 2 * kh + 0);      // K = 0 / 2
    A0[1] = ec_wload(W, m, 2 * kh + 1);      // K = 1 / 3
    A1[0] = ec_wload(W, m, 4 + 2 * kh + 0);  // K = 4 / 6
    A1[1] = ec_wload(W, m, 4 + 2 * kh + 1);  // K = 5 / 7(=0)

    // --- C = bias broadcast: VGPR r holds M=r (lanes<16) / M=8+r (lanes>=16) ---
    v8f Cb = {};
#pragma unroll
    for (int r = 0; r < 7; ++r) Cb[r] = (lane < 16) ? b[r] : 0.0f;

    // --- two 16-edge tiles per wave ---
#pragma unroll
    for (int t = 0; t < 2; ++t) {
        const int srcL = t * 16 + (lane & 15);   // lane holding edge for column N
        float lo, hi;
        v2f B0, B1;
        lo = __shfl(f0, srcL); hi = __shfl(f2, srcL); B0[0] = (lane < 16) ? lo : hi;
        lo = __shfl(f1, srcL); hi = __shfl(f3, srcL); B0[1] = (lane < 16) ? lo : hi;
        lo = __shfl(f4, srcL); hi = __shfl(f6, srcL); B1[0] = (lane < 16) ? lo : hi;
        lo = __shfl(f5, srcL); hi = __shfl(f7, srcL); B1[1] = (lane < 16) ? lo : hi;

        v8f acc = Cb;
        acc = __builtin_amdgcn_wmma_f32_16x16x4_f32(false, A0, false, B0,
                                                    (short)0, acc, false, false);
        acc = __builtin_amdgcn_wmma_f32_16x16x4_f32(false, A1, false, B1,
                                                    (short)0, acc, false, false);

        // scatter: lane L<16 holds all 7 outputs of edge (base + t*16 + L)
        const int  dd   = __shfl(d, srcL);
        const bool eact = (base + t * 16 + (lane & 15)) < E;
        if (lane < 16 && eact) {
            float* o = out + (long)dd * 7;
#pragma unroll
            for (int r = 0; r < 7; ++r) {
                unsafeAtomicAdd(o + r, acc[r]);   // global_atomic_add_f32 (no return)
            }
        }
    }
}

// ---------------------------------------------------------------------------
extern "C" void kernel_launch(void* const* d_in, const int* in_sizes, int n_in,
                              void* d_out, int out_size, void* d_ws, size_t ws_size,
                              hipStream_t stream) {
    const float* h    = (const float*)d_in[0];   // [N,3]
    const int*   esrc = (const int*)d_in[1];     // [E]
    const int*   edst = (const int*)d_in[2];     // [E]
    const float* W    = (const float*)d_in[3];   // [7,7]
    const float* b    = (const float*)d_in[4];   // [7]
    float* out = (float*)d_out;                  // [N,7]
    float* ws  = (float*)d_ws;

    const int E = in_sizes[1];

    // 1) zero output accumulator
    {
        int blocks = (out_size + 255) / 256;
        ec_zero_kernel<<<blocks, 256, 0, stream>>>(out, out_size);
    }
    // 2) deterministic mean of h[edge_dst]
    ec_mean_partial_kernel<<<NB_MEAN, 256, 0, stream>>>(h, edst, ws + WS_PARTIALS, E);
    ec_mean_final_kernel<<<1, 256, 0, stream>>>(ws + WS_PARTIALS, ws + WS_SUMS, NB_MEAN);
    // 3) fused edge features + WMMA linear + scatter-add
    {
        long waves  = ((long)E + 31) / 32;
        long blocks = (waves + 7) / 8;           // 256 threads = 8 waves
        ec_edge_wmma_kernel<<<(int)blocks, 256, 0, stream>>>(h, esrc, edst, W, b,
                                                             ws + WS_SUMS, out, E);
    }
}